// GAE_GIN_36704790512253
// MI455X (gfx1250) — compile-verified
//
#include <hip/hip_runtime.h>

#define N_NODES   100000
#define N_EDGES   1600000
#define HID       128
#define EMB       256
#define NGRAPHS   512
#define BN_EPS    1e-5f

typedef __attribute__((ext_vector_type(2))) float v2f;
typedef __attribute__((ext_vector_type(8))) float v8f;
typedef __attribute__((ext_vector_type(4))) int   v4i_;

// ---------------------------------------------------------------------------
// Async global->LDS staging (gfx1250 GLOBAL_LOAD_ASYNC_TO_LDS_B128, ASYNCcnt),
// with a synchronous float4 fallback if the builtin is unavailable.
// Builtin signature (from hipcc diagnostic): param0 = v4i* in __device__ (AS1).
// ---------------------------------------------------------------------------
#if defined(__has_builtin)
#if __has_builtin(__builtin_amdgcn_global_load_async_to_lds_b128)
#define HAVE_ASYNC_LDS 1
#endif
#endif

__device__ __forceinline__ void stage_f4(const float* g, float* l) {
#ifdef HAVE_ASYNC_LDS
  __builtin_amdgcn_global_load_async_to_lds_b128(
      (__attribute__((address_space(1))) v4i_*)g,
      (__attribute__((address_space(3))) v4i_*)l, 0, 0);
#else
  *(float4*)l = *(const float4*)g;
#endif
}

__device__ __forceinline__ void stage_wait_barrier() {
#ifdef HAVE_ASYNC_LDS
  asm volatile("s_wait_asynccnt 0x0" ::: "memory");
#endif
  __syncthreads();
}

// ---------------------------------------------------------------------------
// z[n,0:128] = h[n*ldh + c]   (GIN self term; scatter adds neighbors on top)
// ---------------------------------------------------------------------------
__global__ void init_z(float* __restrict__ z, const float* __restrict__ h,
                       int ldh, int nnodes) {
  long long t = (long long)blockIdx.x * blockDim.x + threadIdx.x;
  if (t >= (long long)nnodes * HID) return;
  int n = (int)(t >> 7);
  int c = (int)(t & 127);
  z[t] = h[(long long)n * ldh + c];
}

// ---------------------------------------------------------------------------
// One wave per edge: coalesced float4 row gather + 128 contiguous f32 atomics
// ---------------------------------------------------------------------------
__global__ void scatter_edges(const float* __restrict__ h, int ldh,
                              const long long* __restrict__ src,
                              const long long* __restrict__ dst,
                              float* __restrict__ z, int nedges) {
  long long t = (long long)blockIdx.x * blockDim.x + threadIdx.x;
  int e = (int)(t >> 5);
  int lane = (int)(t & 31);
  if (e >= nedges) return;
  long long s = src[e];
  long long d = dst[e];
  const float4 hv = ((const float4*)(h + s * ldh))[lane];
  float* zp = z + d * HID + lane * 4;
  atomicAdd(zp + 0, hv.x);
  atomicAdd(zp + 1, hv.y);
  atomicAdd(zp + 2, hv.z);
  atomicAdd(zp + 3, hv.w);
}

// ---------------------------------------------------------------------------
// C[M,N] = act(A[M,K] @ W[K,N] + bias) (+res)  via v_wmma_f32_16x16x4_f32.
// BR rows/block (BR/16 waves, each wave a 16-row strip). A staged once in LDS
// (async-to-LDS when available), W staged per 16-col tile; next W tile
// prefetched (global_prefetch_b8) during compute.
// ---------------------------------------------------------------------------
template <int K, int BR, bool RELU, bool HASRES>
__global__ void gemm_wmma(const float* __restrict__ A, int lda,
                          const float* __restrict__ W,
                          const float* __restrict__ bias,
                          float* __restrict__ C, int ldc,
                          int M, int N,
                          const float* __restrict__ res) {
  constexpr int NWAVES = BR / 16;
  constexpr int NTHREADS = NWAVES * 32;
  constexpr int LDSW = K + 4;                 // +4 pad spreads A rows over banks
  __shared__ float sA[BR * LDSW];
  __shared__ float sW[K * 16];

  const int tid = threadIdx.x;
  const int lane = tid & 31;
  const int wave = tid >> 5;
  const int row0 = blockIdx.x * BR;
  const bool fullblk = (row0 + BR) <= M;

  // ---- stage A[row0:row0+BR, 0:K] ----
  constexpr int RCH = K / 4;                  // float4 chunks per row
  if (fullblk) {
    for (int q = tid; q < BR * RCH; q += NTHREADS) {
      int r = q / RCH;
      int k4 = (q - r * RCH) * 4;
      stage_f4(A + (long long)(row0 + r) * lda + k4, &sA[r * LDSW + k4]);
    }
  } else {
    for (int q = tid; q < BR * RCH; q += NTHREADS) {
      int r = q / RCH;
      int k4 = (q - r * RCH) * 4;
      float4 v = make_float4(0.f, 0.f, 0.f, 0.f);
      if (row0 + r < M) v = *(const float4*)(A + (long long)(row0 + r) * lda + k4);
      *(float4*)&sA[r * LDSW + k4] = v;       // (K+4) % 4 == 0 -> 16B aligned
    }
  }

  const int wrow = wave * 16;
  const int mlane = lane & 15;                // M (A,C) / N (B,C) sub-index
  const int khalf = (lane >> 4) << 1;         // K pair selector per lane half
  const int ntiles = N >> 4;
  const int arow = (wrow + mlane) * LDSW;

  for (int nt = 0; nt < ntiles; ++nt) {
    // ---- stage W[0:K, nt*16 .. +16) into sW[k*16 + c] (coalesced b128) ----
    for (int q = tid; q < K * 4; q += NTHREADS) {
      int k = q >> 2;
      int c4 = (q & 3) << 2;
      stage_f4(W + (long long)k * N + (nt << 4) + c4, &sW[k * 16 + c4]);
    }
    if (nt + 1 < ntiles)                      // prefetch next W tile
      __builtin_prefetch(W + (long long)(tid & (K - 1)) * N + ((nt + 1) << 4), 0, 3);
    stage_wait_barrier();                     // s_wait_asynccnt + barrier

    v8f acc = {0.f, 0.f, 0.f, 0.f, 0.f, 0.f, 0.f, 0.f};
    for (int k0 = 0; k0 < K; k0 += 4) {
      v2f a, b;
      a.x = sA[arow + k0 + khalf];
      a.y = sA[arow + k0 + khalf + 1];
      b.x = sW[(k0 + khalf) * 16 + mlane];
      b.y = sW[(k0 + khalf + 1) * 16 + mlane];
      acc = __builtin_amdgcn_wmma_f32_16x16x4_f32(
          false, a, false, b, (short)0, acc, false, false);
    }

    const int col = (nt << 4) + mlane;
    const float bv = bias[col];
#pragma unroll
    for (int v = 0; v < 8; ++v) {
      int grow = row0 + wrow + v + ((lane >> 4) << 3);
      float val = acc[v] + bv;
      if (RELU) val = fmaxf(val, 0.f);
      if (grow < M) {
        long long idx = (long long)grow * ldc + col;
        if (HASRES) val += res[idx];
        C[idx] = val;
      }
    }
    __syncthreads();                          // protect sW before next tile
  }
}

// ---------------------------------------------------------------------------
__global__ void zero_buf(float* __restrict__ p, int n) {
  int t = blockIdx.x * blockDim.x + threadIdx.x;
  if (t < n) p[t] = 0.f;
}

// per-channel sum / sumsq (128 channels), atomics once per block per channel
__global__ void bn_stats(const float* __restrict__ z, float* __restrict__ stats,
                         int nnodes) {
  const int c = threadIdx.x;            // 128 threads
  int r0 = blockIdx.x * 512;
  int r1 = min(nnodes, r0 + 512);
  float s = 0.f, sq = 0.f;
  for (int r = r0; r < r1; ++r) {
    float v = z[(long long)r * HID + c];
    s += v;
    sq += v * v;
  }
  atomicAdd(&stats[c], s);
  atomicAdd(&stats[HID + c], sq);
}

__global__ void bn_finalize(const float* __restrict__ stats,
                            const float* __restrict__ gamma,
                            const float* __restrict__ beta,
                            float* __restrict__ ss, int nnodes) {
  int c = threadIdx.x;                  // 128 threads
  if (c >= HID) return;
  float inv_n = 1.f / (float)nnodes;
  float mean = stats[c] * inv_n;
  float var = stats[HID + c] * inv_n - mean * mean;   // biased, matches jnp.var
  float scale = gamma[c] * rsqrtf(var + BN_EPS);
  ss[c] = scale;
  ss[HID + c] = beta[c] - mean * scale;
}

// write normalized h into local_vector column slice (row stride 256)
__global__ void bn_apply(const float* __restrict__ z, const float* __restrict__ ss,
                         float* __restrict__ dst, int nnodes) {
  long long t = (long long)blockIdx.x * blockDim.x + threadIdx.x;
  if (t >= (long long)nnodes * HID) return;
  int n = (int)(t >> 7);
  int c = (int)(t & 127);
  dst[(long long)n * EMB + c] = z[t] * ss[c] + ss[HID + c];
}

// segment-sum pooling over sorted batch: running sum, atomic only at boundaries
__global__ void pool_graphs(const float* __restrict__ lv,
                            const long long* __restrict__ batch,
                            float* __restrict__ gvec, int nnodes) {
  const int c = threadIdx.x;            // 256 threads = channels
  int r0 = blockIdx.x * 256;
  int r1 = min(nnodes, r0 + 256);
  if (r0 >= r1) return;
  long long g = batch[r0];
  float sum = 0.f;
  for (int r = r0; r < r1; ++r) {
    long long gr = batch[r];
    if (gr != g) {
      atomicAdd(&gvec[g * EMB + c], sum);
      sum = 0.f;
      g = gr;
    }
    sum += lv[(long long)r * EMB + c];
  }
  atomicAdd(&gvec[g * EMB + c], sum);
}

// ---------------------------------------------------------------------------
extern "C" void kernel_launch(void* const* d_in, const int* in_sizes, int n_in,
                              void* d_out, int out_size, void* d_ws, size_t ws_size,
                              hipStream_t stream) {
  const float* x        = (const float*)d_in[0];
  const long long* ei   = (const long long*)d_in[1];   // int64 [2, NE]
  const long long* batch= (const long long*)d_in[2];   // int64 [N]
  const float* convW1   = (const float*)d_in[3];
  const float* convb1   = (const float*)d_in[4];
  const float* convW2   = (const float*)d_in[5];
  const float* convb2   = (const float*)d_in[6];
  const float* bn_gamma = (const float*)d_in[7];
  const float* bn_beta  = (const float*)d_in[8];
  const float* gW       = (const float*)d_in[9];
  const float* gb       = (const float*)d_in[10];
  const float* g_scW    = (const float*)d_in[11];
  const float* g_scb    = (const float*)d_in[12];
  const float* lW       = (const float*)d_in[13];
  const float* lb       = (const float*)d_in[14];
  const float* l_scW    = (const float*)d_in[15];
  const float* l_scb    = (const float*)d_in[16];

  const long long* src = ei;
  const long long* dst = ei + N_EDGES;

  float* out = (float*)d_out;
  float* og = out;                         // global_out [512,256]
  float* ol = out + (size_t)NGRAPHS * EMB; // local_out  [N,256]

  // workspace layout
  float* wsf = (float*)d_ws;
  size_t off = 0;
  float* lv   = wsf + off; off += (size_t)N_NODES * EMB;   // local_vector
  float* t0   = wsf + off; off += (size_t)N_NODES * EMB;   // z|y1 then FF hidden
  float* zbuf = t0;
  float* ybuf = t0 + (size_t)N_NODES * HID;
  float* stats = wsf + off; off += 2 * HID;
  float* ss    = wsf + off; off += 2 * HID;
  float* gvec  = wsf + off; off += (size_t)NGRAPHS * EMB;
  float* gt0   = wsf + off; off += (size_t)NGRAPHS * EMB;
  float* gt1   = wsf + off; off += (size_t)NGRAPHS * EMB;

  const int nh_blocks  = (int)(((long long)N_NODES * HID + 255) / 256);
  const int ed_blocks  = (int)(((long long)N_EDGES * 32 + 255) / 256);
  const int g128_blk   = (N_NODES + 63) / 64;   // gemm<128,64,...>
  const int g256_blk   = (N_NODES + 31) / 32;   // gemm<256,32,...>
  const int gg256_blk  = (NGRAPHS + 31) / 32;

  // -------- GIN layers --------
  for (int layer = 0; layer < 2; ++layer) {
    const float* h = (layer == 0) ? x : lv;   // layer1 input = xs[0] at lv cols [0,128)
    const int ldh = (layer == 0) ? HID : EMB;

    init_z<<<nh_blocks, 256, 0, stream>>>(zbuf, h, ldh, N_NODES);
    scatter_edges<<<ed_blocks, 256, 0, stream>>>(h, ldh, src, dst, zbuf, N_EDGES);

    gemm_wmma<128, 64, true, false><<<g128_blk, 128, 0, stream>>>(
        zbuf, HID, convW1 + (size_t)layer * HID * HID, convb1 + layer * HID,
        ybuf, HID, N_NODES, HID, nullptr);
    gemm_wmma<128, 64, true, false><<<g128_blk, 128, 0, stream>>>(
        ybuf, HID, convW2 + (size_t)layer * HID * HID, convb2 + layer * HID,
        zbuf, HID, N_NODES, HID, nullptr);

    zero_buf<<<1, 256, 0, stream>>>(stats, 2 * HID);
    bn_stats<<<(N_NODES + 511) / 512, 128, 0, stream>>>(zbuf, stats, N_NODES);
    bn_finalize<<<1, 128, 0, stream>>>(stats, bn_gamma + layer * HID,
                                       bn_beta + layer * HID, ss, N_NODES);
    bn_apply<<<nh_blocks, 256, 0, stream>>>(zbuf, ss, lv + layer * HID, N_NODES);
  }

  // -------- pooling --------
  zero_buf<<<(NGRAPHS * EMB + 255) / 256, 256, 0, stream>>>(gvec, NGRAPHS * EMB);
  pool_graphs<<<(N_NODES + 255) / 256, 256, 0, stream>>>(lv, batch, gvec, N_NODES);

  // -------- global FF: out = relu-chain(gvec) + gvec @ scW + scb --------
  gemm_wmma<256, 32, false, false><<<gg256_blk, 64, 0, stream>>>(
      gvec, EMB, g_scW, g_scb, og, EMB, NGRAPHS, EMB, nullptr);
  gemm_wmma<256, 32, true, false><<<gg256_blk, 64, 0, stream>>>(
      gvec, EMB, gW, gb, gt0, EMB, NGRAPHS, EMB, nullptr);
  gemm_wmma<256, 32, true, false><<<gg256_blk, 64, 0, stream>>>(
      gt0, EMB, gW + (size_t)EMB * EMB, gb + EMB, gt1, EMB, NGRAPHS, EMB, nullptr);
  gemm_wmma<256, 32, true, true><<<gg256_blk, 64, 0, stream>>>(
      gt1, EMB, gW + (size_t)2 * EMB * EMB, gb + 2 * EMB, og, EMB, NGRAPHS, EMB, og);

  // -------- local FF (shortcut first so lv can be recycled) --------
  gemm_wmma<256, 32, false, false><<<g256_blk, 64, 0, stream>>>(
      lv, EMB, l_scW, l_scb, ol, EMB, N_NODES, EMB, nullptr);
  gemm_wmma<256, 32, true, false><<<g256_blk, 64, 0, stream>>>(
      lv, EMB, lW, lb, t0, EMB, N_NODES, EMB, nullptr);
  gemm_wmma<256, 32, true, false><<<g256_blk, 64, 0, stream>>>(
      t0, EMB, lW + (size_t)EMB * EMB, lb + EMB, lv, EMB, N_NODES, EMB, nullptr);
  gemm_wmma<256, 32, true, true><<<g256_blk, 64, 0, stream>>>(
      lv, EMB, lW + (size_t)2 * EMB * EMB, lb + 2 * EMB, ol, EMB, N_NODES, EMB, ol);
}